// GCN_38053410242794
// MI455X (gfx1250) — compile-verified
//
#include <hip/hip_runtime.h>
#include <hip/hip_bf16.h>
#include <math.h>

typedef __attribute__((ext_vector_type(2))) float v2f;
typedef __attribute__((ext_vector_type(8))) float v8f;

// ---------------------------------------------------------------------------
// small utility kernels
// ---------------------------------------------------------------------------
__global__ void k_fill(float* p, float v, int n) {
  int i = blockIdx.x * blockDim.x + threadIdx.x;
  if (i < n) p[i] = v;
}

__global__ void k_deg_count(const int* __restrict__ dst, float* deg, int E) {
  int i = blockIdx.x * blockDim.x + threadIdx.x;
  if (i < E) atomicAdd(&deg[dst[i]], 1.0f);
}

__global__ void k_rsqrt_inplace(float* p, int n) {
  int i = blockIdx.x * blockDim.x + threadIdx.x;
  if (i < n) p[i] = rsqrtf(p[i]);
}

// h1[v,f] = x[v] * W1[f]   (W1 is [1,64])
__global__ void k_h1(const float* __restrict__ x, const float* __restrict__ W1,
                     float* __restrict__ h1, int n) {
  int i = blockIdx.x * blockDim.x + threadIdx.x;
  if (i < n * 64) {
    int v = i >> 6;
    int f = i & 63;
    h1[i] = x[v] * W1[f];
  }
}

// agg[v,f] = dinv[v]^2 * h[v,f]   (self-loop term; also initializes agg)
__global__ void k_selfloop(const float* __restrict__ dinv, const float* __restrict__ h,
                           float* __restrict__ agg, int n, int logF) {
  size_t i = (size_t)blockIdx.x * blockDim.x + threadIdx.x;
  size_t total = (size_t)n << logF;
  if (i < total) {
    int v = (int)(i >> logF);
    float dv = dinv[v];
    agg[i] = dv * dv * h[i];
  }
}

// for each edge e and 4-feature chunk c:  agg[dst, 4c..4c+3] += norm * h[src, 4c..4c+3]
__global__ void k_edge_agg(const int* __restrict__ src, const int* __restrict__ dst,
                           const float* __restrict__ dinv, const float* __restrict__ h,
                           float* agg, int E, int logc, int F) {
  size_t tid = (size_t)blockIdx.x * blockDim.x + threadIdx.x;
  int e = (int)(tid >> logc);
  if (e >= E) return;
  int c = (int)(tid & ((1u << logc) - 1u));
  int s = src[e];
  int d = dst[e];
  float nrm = dinv[s] * dinv[d];
  const float4 hv = *reinterpret_cast<const float4*>(h + (size_t)s * F + c * 4);
  float* ap = agg + (size_t)d * F + c * 4;
  atomicAdd(ap + 0, nrm * hv.x);
  atomicAdd(ap + 1, nrm * hv.y);
  atomicAdd(ap + 2, nrm * hv.z);
  atomicAdd(ap + 3, nrm * hv.w);
}

// a[i] = relu(a[i] + bias[i % F]),  F power of two
__global__ void k_bias_relu(float* a, const float* __restrict__ b, size_t total, int Fmask) {
  size_t i = (size_t)blockIdx.x * blockDim.x + threadIdx.x;
  if (i < total) a[i] = fmaxf(a[i] + b[(int)(i & (size_t)Fmask)], 0.0f);
}

// ---------------------------------------------------------------------------
// f32 WMMA GEMM:  C[M,N] = A[M,K] @ B[K,N]   (K = 64, N multiple of 16)
// grid.x = ceil(M/16), 256 threads = 8 waves; wave w owns N-tile w (N=128).
// Uses V_WMMA_F32_16X16X4_F32 — full-precision fp32 matrix op.
// ---------------------------------------------------------------------------
__global__ void k_gemm_f32_wmma(const float* __restrict__ A, const float* __restrict__ B,
                                float* __restrict__ C, int M, int N, int K) {
  int lane = threadIdx.x & 31;
  int wave = threadIdx.x >> 5;
  int n0 = wave * 16;
  if (n0 >= N) return;  // never diverges within a wave
  int m0 = blockIdx.x * 16;
  int half = lane >> 4;  // 0: K pair (k0,k0+1) ; 1: K pair (k0+2,k0+3)
  int l = lane & 15;
  int row = m0 + l;
  bool rowOK = row < M;
  size_t aBase = (size_t)(rowOK ? row : m0) * (size_t)K;  // clamped, always valid
  int col = n0 + l;

  v8f acc = {};
#pragma unroll
  for (int k0 = 0; k0 < 64; k0 += 4) {
    int ka = k0 + half * 2;
    v2f a = *reinterpret_cast<const v2f*>(A + aBase + ka);  // 8B aligned (ka even)
    if (!rowOK) { a.x = 0.0f; a.y = 0.0f; }
    v2f bb;
    bb.x = B[(size_t)(ka + 0) * N + col];
    bb.y = B[(size_t)(ka + 1) * N + col];
    acc = __builtin_amdgcn_wmma_f32_16x16x4_f32(
        /*neg_a=*/false, a, /*neg_b=*/false, bb,
        /*c_mod=*/(short)0, acc, /*reuse_a=*/false, /*reuse_b=*/false);
  }

  int rBase = m0 + half * 8;  // VGPR r holds row rBase + r, column col
#pragma unroll
  for (int r = 0; r < 8; ++r) {
    int rr = rBase + r;
    if (rr < M) C[(size_t)rr * N + col] = acc[r];
  }
}

// ---------------------------------------------------------------------------
// pooling + head
// ---------------------------------------------------------------------------
__global__ void k_pool_sum(const float* __restrict__ h, const int* __restrict__ batch,
                           float* sums, float* cnts, int n) {
  size_t i = (size_t)blockIdx.x * blockDim.x + threadIdx.x;
  size_t total = (size_t)n << 7;  // F = 128
  if (i < total) {
    int v = (int)(i >> 7);
    int f = (int)(i & 127);
    int g = batch[v];
    atomicAdd(&sums[(size_t)g * 128 + f], h[i]);
    if (f == 0) atomicAdd(&cnts[g], 1.0f);
  }
}

__global__ void k_pool_div(float* sums, const float* __restrict__ cnts) {
  int i = blockIdx.x * blockDim.x + threadIdx.x;  // 512*128
  if (i < 512 * 128) {
    int g = i >> 7;
    sums[i] = sums[i] / fmaxf(cnts[g], 1.0f);
  }
}

// out[g,j] = relu( sum_k pooled[g,k]*W[k,j] + b[j] )  ; [512,128]@[128,64]
__global__ void k_fc1(const float* __restrict__ pooled, const float* __restrict__ W,
                      const float* __restrict__ b, float* __restrict__ out) {
  int g = blockIdx.x;
  int j = threadIdx.x;  // 64 threads
  float s = b[j];
  const float* p = pooled + (size_t)g * 128;
#pragma unroll 8
  for (int k = 0; k < 128; ++k) s += p[k] * W[(size_t)k * 64 + j];
  out[(size_t)g * 64 + j] = fmaxf(s, 0.0f);
}

// logits = h@W + b ; out = log_softmax(logits)   [512,64]@[64,10]
__global__ void k_fc2_logsoftmax(const float* __restrict__ h, const float* __restrict__ W,
                                 const float* __restrict__ b, float* __restrict__ out) {
  int g = blockIdx.x * blockDim.x + threadIdx.x;  // one thread per graph
  if (g >= 512) return;
  const float* hg = h + (size_t)g * 64;
  float logit[10];
#pragma unroll
  for (int j = 0; j < 10; ++j) logit[j] = b[j];
  for (int k = 0; k < 64; ++k) {
    float hv = hg[k];
#pragma unroll
    for (int j = 0; j < 10; ++j) logit[j] += hv * W[k * 10 + j];
  }
  float mx = logit[0];
#pragma unroll
  for (int j = 1; j < 10; ++j) mx = fmaxf(mx, logit[j]);
  float sum = 0.0f;
#pragma unroll
  for (int j = 0; j < 10; ++j) sum += expf(logit[j] - mx);
  float lse = mx + logf(sum);
  float* og = out + (size_t)g * 10;
#pragma unroll
  for (int j = 0; j < 10; ++j) og[j] = logit[j] - lse;
}

// ---------------------------------------------------------------------------
// launch
// ---------------------------------------------------------------------------
extern "C" void kernel_launch(void* const* d_in, const int* in_sizes, int n_in,
                              void* d_out, int out_size, void* d_ws, size_t ws_size,
                              hipStream_t stream) {
  const float* x    = (const float*)d_in[0];
  const float* W1   = (const float*)d_in[1];
  const float* b1   = (const float*)d_in[2];
  const float* W2   = (const float*)d_in[3];
  const float* b2   = (const float*)d_in[4];
  const float* fcW1 = (const float*)d_in[5];
  const float* fcb1 = (const float*)d_in[6];
  const float* fcW2 = (const float*)d_in[7];
  const float* fcb2 = (const float*)d_in[8];
  const int*   edge = (const int*)d_in[9];
  const int*   batch= (const int*)d_in[10];

  const int n = in_sizes[0];        // 100000 nodes
  const int E = in_sizes[9] / 2;    // 640000 edges
  const int* src = edge;
  const int* dst = edge + E;

  // workspace layout (floats), 256B-aligned chunks
  float* ws = (float*)d_ws;
  size_t off = 0;
  auto alloc = [&](size_t cnt) { float* p = ws + off; off += (cnt + 63) & ~(size_t)63; return p; };
  float* dinv  = alloc((size_t)n);            // deg -> rsqrt in place
  float* h1    = alloc((size_t)n * 64);       // x@W1
  float* agg1  = alloc((size_t)n * 64);       // conv1 accum -> relu (h1r)
  float* h2    = alloc((size_t)n * 128);      // h1r@W2 (WMMA GEMM out)
  float* agg2  = alloc((size_t)n * 128);      // conv2 accum -> relu (h2r)
  float* sums  = alloc((size_t)512 * 128);    // pooled
  float* cnts  = alloc((size_t)512);
  float* fc1o  = alloc((size_t)512 * 64);

  const int B = 256;
  auto blocks = [](size_t work, int blk) { return (unsigned)((work + blk - 1) / blk); };

  // degree with self-loops, then dinv = rsqrt(deg)
  k_fill<<<blocks(n, B), B, 0, stream>>>(dinv, 1.0f, n);
  k_deg_count<<<blocks(E, B), B, 0, stream>>>(dst, dinv, E);
  k_rsqrt_inplace<<<blocks(n, B), B, 0, stream>>>(dinv, n);

  // layer 1: lin -> aggregate -> bias+relu
  k_h1<<<blocks((size_t)n * 64, B), B, 0, stream>>>(x, W1, h1, n);
  k_selfloop<<<blocks((size_t)n * 64, B), B, 0, stream>>>(dinv, h1, agg1, n, 6);
  k_edge_agg<<<blocks((size_t)E * 16, B), B, 0, stream>>>(src, dst, dinv, h1, agg1, E, 4, 64);
  k_bias_relu<<<blocks((size_t)n * 64, B), B, 0, stream>>>(agg1, b1, (size_t)n * 64, 63);

  // layer 2: WMMA f32 GEMM [n,64]@[64,128] -> aggregate -> bias+relu
  k_gemm_f32_wmma<<<blocks((size_t)n, 16), 256, 0, stream>>>(agg1, W2, h2, n, 128, 64);
  k_selfloop<<<blocks((size_t)n * 128, B), B, 0, stream>>>(dinv, h2, agg2, n, 7);
  k_edge_agg<<<blocks((size_t)E * 32, B), B, 0, stream>>>(src, dst, dinv, h2, agg2, E, 5, 128);
  k_bias_relu<<<blocks((size_t)n * 128, B), B, 0, stream>>>(agg2, b2, (size_t)n * 128, 127);

  // mean pool per graph
  k_fill<<<blocks(512 * 128, B), B, 0, stream>>>(sums, 0.0f, 512 * 128);
  k_fill<<<blocks(512, B), B, 0, stream>>>(cnts, 0.0f, 512);
  k_pool_sum<<<blocks((size_t)n * 128, B), B, 0, stream>>>(agg2, batch, sums, cnts, n);
  k_pool_div<<<blocks(512 * 128, B), B, 0, stream>>>(sums, cnts);

  // FC head + log_softmax
  k_fc1<<<512, 64, 0, stream>>>(sums, fcW1, fcb1, fc1o);
  k_fc2_logsoftmax<<<2, 256, 0, stream>>>(fc1o, fcW2, fcb2, (float*)d_out);
}